// SparseConvolutionBase_86595130622318
// MI455X (gfx1250) — compile-verified
//
#include <hip/hip_runtime.h>

typedef __attribute__((ext_vector_type(2))) float v2f;
typedef __attribute__((ext_vector_type(8))) float v8f;

#define C_DIM 64   // C_IN == C_OUT == 64
#define KOFF  27   // 3^3 kernel offsets

// ---------------------------------------------------------------------------
// Zero-fill for d_out (harness poisons it with 0xAA before timing).
// ---------------------------------------------------------------------------
__global__ void __launch_bounds__(256) sconv_zero_kernel(float* __restrict__ p, int n) {
    int i = blockIdx.x * blockDim.x + threadIdx.x;
    if (i < n) p[i] = 0.0f;
}

// ---------------------------------------------------------------------------
// Sparse conv: gather -> fp32 WMMA GEMM -> scatter-add.
// grid = (ceil(ntiles/8), 27), block = 256 threads (8 waves, wave32).
// Each wave computes one 16x64 output tile for kernel offset k = blockIdx.y.
// ---------------------------------------------------------------------------
__global__ void __launch_bounds__(256) sconv_wmma_f32_kernel(
    const float* __restrict__ in_feats,   // [N_IN, 64]
    const float* __restrict__ kernel,     // [27, 64, 64] (C_IN major)
    const int*   __restrict__ in_map,     // [27, M]
    const int*   __restrict__ out_map,    // [27, M]
    float*       __restrict__ out,        // [N_OUT, 64]
    int M, int ntiles)
{
    __shared__ float sW[C_DIM * C_DIM];   // 16 KB: W[k], shared by 8 waves

    const int k = blockIdx.y;

    // ---- Stage W[k] (64x64 fp32) into LDS via CDNA5 async copy:
    // global_load_async_to_lds_b128 copies cache->LDS without VGPR round-trip,
    // tracked by ASYNCcnt (no s_wait_loadcnt serialization in the prologue).
    {
        const float* wk = kernel + (size_t)k * C_DIM * C_DIM;
        // Low 32 bits of a generic LDS pointer == LDS byte offset (aperture map).
        unsigned lds_base = (unsigned)(size_t)(void*)sW;
        #pragma unroll
        for (int it = 0; it < 4; ++it) {
            int f4 = threadIdx.x + it * 256;            // float4 index, 1024 total
            unsigned lds_addr = lds_base + (unsigned)(f4 * 16);
            const void* gaddr = (const void*)(wk + f4 * 4);
            asm volatile("global_load_async_to_lds_b128 %0, %1, off"
                         :: "v"(lds_addr), "v"(gaddr)
                         : "memory");
        }
        asm volatile("s_wait_asynccnt 0x0" ::: "memory");
    }
    __syncthreads();

    const int wave = threadIdx.x >> 5;
    const int lane = threadIdx.x & 31;
    const int half = lane >> 4;          // 0: lanes 0-15, 1: lanes 16-31
    const int col  = lane & 15;

    const int tile = blockIdx.x * 8 + wave;   // wave-uniform
    if (tile >= ntiles) return;               // uniform exit -> EXEC stays all-1s
    const int m0 = tile * 16;

    // ---- Gather A panel (16 rows x 64 cols fp32) into WMMA A-fragments.
    // A 16x4 f32 layout: lanes hold M = lane&15; VGPR0 = K=2h, VGPR1 = K=2h+1.
    int mi = m0 + col;
    if (mi >= M) mi = M - 1;                  // clamp (scatter is predicated)
    const int row = in_map[(size_t)k * M + mi];
    const float* arow = in_feats + (size_t)row * C_DIM + 2 * half;
    v2f a[16];
    #pragma unroll
    for (int c = 0; c < 16; ++c)
        a[c] = *(const v2f*)(arow + 4 * c);   // global_load_b64, L2-resident

    // ---- Output row indices for this lane's D fragment.
    // D 16x16 f32 layout: VGPR r holds M = r + 8*half, N = lane&15.
    int  orow[8];
    bool oval[8];
    const int* omk = out_map + (size_t)k * M + m0 + 8 * half;
    #pragma unroll
    for (int r = 0; r < 8; ++r) {
        int m   = m0 + 8 * half + r;
        oval[r] = (m < M);
        orow[r] = oval[r] ? omk[r] : 0;
    }

    // ---- GEMM: D[16x64] = A[16x64] x W[64x64], in 4 N-tiles of 16 cols.
    #pragma unroll
    for (int n = 0; n < 4; ++n) {
        v8f acc = {};
        // B 4x16 f32 layout: VGPR0 = K=2h, VGPR1 = K=2h+1, N = lane&15.
        const float* b0 = sW + (2 * half) * C_DIM + n * 16 + col;
        #pragma unroll
        for (int c = 0; c < 16; ++c) {
            v2f b;
            b.x = b0[c * 4 * C_DIM];            // ds_load row 4c+2h
            b.y = b0[c * 4 * C_DIM + C_DIM];    // ds_load row 4c+2h+1
            acc = __builtin_amdgcn_wmma_f32_16x16x4_f32(
                      /*neg_a=*/false, a[c],
                      /*neg_b=*/false, b,
                      /*c_mod=*/(short)0, acc,
                      /*reuse_a=*/false, /*reuse_b=*/false);
        }
        // ---- Scatter-add into out rows (hardware fp32 atomics, L2-resident).
        float* obase = out + n * 16 + col;
        #pragma unroll
        for (int r = 0; r < 8; ++r) {
            if (oval[r])
                unsafeAtomicAdd(obase + (size_t)orow[r] * C_DIM, acc[r]);
        }
    }
}

// ---------------------------------------------------------------------------
extern "C" void kernel_launch(void* const* d_in, const int* in_sizes, int n_in,
                              void* d_out, int out_size, void* d_ws, size_t ws_size,
                              hipStream_t stream) {
    const float* in_feats = (const float*)d_in[0];
    const float* kern     = (const float*)d_in[1];
    const int*   in_map   = (const int*)d_in[2];
    const int*   out_map  = (const int*)d_in[3];
    float*       out      = (float*)d_out;

    const int M      = in_sizes[2] / KOFF;      // 150000
    const int ntiles = (M + 15) / 16;           // 9375

    // Zero output (stream-ordered before the conv kernel).
    {
        int blocks = (out_size + 255) / 256;
        sconv_zero_kernel<<<blocks, 256, 0, stream>>>(out, out_size);
    }

    dim3 grid((ntiles + 7) / 8, KOFF);
    sconv_wmma_f32_kernel<<<grid, 256, 0, stream>>>(
        in_feats, kern, in_map, out_map, out, M, ntiles);
}